// DiGAMN_Model_34179349742161
// MI455X (gfx1250) — compile-verified
//
#include <hip/hip_runtime.h>
#include <hip/hip_bf16.h>

#define N_NODES 50000
#define N_HID   32
#define N_EDGES (N_NODES * 16)
#define N_RELS  4
#define MEMD    8
#define HEADS   4
#define SLOPE   0.2f

typedef __attribute__((ext_vector_type(2))) float v2f;
typedef __attribute__((ext_vector_type(8))) float v8f;

__device__ __forceinline__ float lrelu(float v) { return v > 0.f ? v : SLOPE * v; }

// Order-preserving float->uint encoding for atomicMax-based segment max.
__device__ __forceinline__ unsigned int fenc(float f) {
    unsigned int u = __float_as_uint(f);
    return (u & 0x80000000u) ? ~u : (u | 0x80000000u);
}
__device__ __forceinline__ float fdec(unsigned int u) {
    return (u & 0x80000000u) ? __uint_as_float(u ^ 0x80000000u) : __uint_as_float(~u);
}

// ---------------------------------------------------------------------------
// Init: x = emb, xcat[:, :32] = emb, deg = 0
// ---------------------------------------------------------------------------
__global__ void global_init_kernel(const float* __restrict__ emb, float* __restrict__ xcat,
                                   float* __restrict__ x, float* __restrict__ deg) {
    int t = blockIdx.x * blockDim.x + threadIdx.x;
    if (t >= N_NODES * N_HID) return;
    int n = t >> 5, o = t & 31;
    float v = emb[t];
    x[t] = v;
    xcat[(size_t)n * 96 + o] = v;
    if (o == 0) deg[n] = 0.f;
}

__global__ void deg_kernel(const int* __restrict__ dst, float* __restrict__ deg) {
    int e = blockIdx.x * blockDim.x + threadIdx.x;
    if (e < N_EDGES) atomicAdd(&deg[dst[e]], 1.0f);
}

// Per-layer init: h = sum_heads bias, agg = 0, emax = enc(-inf-ish), esum = 0
__global__ void layer_init_kernel(const float* __restrict__ b, float* __restrict__ hbuf,
                                  float* __restrict__ agg, unsigned int* __restrict__ emax,
                                  float* __restrict__ esum) {
    int t = blockIdx.x * blockDim.x + threadIdx.x;
    if (t >= N_NODES * N_HID) return;
    int n = t >> 5, o = t & 31;
    hbuf[t] = b[o] + b[32 + o] + b[64 + o] + b[96 + o];
    agg[t] = 0.f;
    if (o < HEADS) { emax[n * HEADS + o] = 0u; esum[n * HEADS + o] = 0.f; }
}

// ---------------------------------------------------------------------------
// fh = x @ W  (50000x32 @ 32x128), fp32 WMMA 16x16x4, one wave per 16x16 tile.
// Block = 256 threads = 8 waves = the full 128 output columns of a 16-row strip.
// ---------------------------------------------------------------------------
__global__ __launch_bounds__(256) void gemm_fh_kernel(const float* __restrict__ X,
                                                      const float* __restrict__ W,
                                                      float* __restrict__ FH) {
    const int wave = threadIdx.x >> 5;
    const int lane = threadIdx.x & 31;
    const int r0 = blockIdx.x * 16;           // 50000/16 = 3125 exact
    const int cb = wave * 16;
    const int m = lane & 15;
    const int koff = (lane >> 4) << 1;        // lanes 16-31 handle K+2/K+3
    v8f c = {};
    const float* xrow = X + (size_t)(r0 + m) * N_HID;
    #pragma unroll
    for (int k0 = 0; k0 < N_HID; k0 += 4) {
        v2f a, b;
        a.x = xrow[k0 + koff];
        a.y = xrow[k0 + koff + 1];
        b.x = W[(size_t)(k0 + koff) * 128 + cb + m];
        b.y = W[(size_t)(k0 + koff + 1) * 128 + cb + m];
        c = __builtin_amdgcn_wmma_f32_16x16x4_f32(false, a, false, b, (short)0, c,
                                                  false, false);
    }
    const int rhalf = (lane < 16) ? 0 : 8;
    #pragma unroll
    for (int j = 0; j < 8; ++j)
        FH[(size_t)(r0 + j + rhalf) * 128 + cb + m] = c[j];
}

// el[n,h] = fh[n,h,:].al[h,:], er likewise
__global__ void eler_kernel(const float* __restrict__ FH, const float* __restrict__ al,
                            const float* __restrict__ ar, float* __restrict__ el,
                            float* __restrict__ er) {
    int t = blockIdx.x * blockDim.x + threadIdx.x;
    if (t >= N_NODES * HEADS) return;
    int n = t >> 2, hh = t & 3;
    const float* f = FH + (size_t)n * 128 + hh * N_HID;
    const float* a = al + hh * N_HID;
    const float* b = ar + hh * N_HID;
    float s1 = 0.f, s2 = 0.f;
    #pragma unroll
    for (int k = 0; k < N_HID; ++k) { s1 += f[k] * a[k]; s2 += f[k] * b[k]; }
    el[t] = s1; er[t] = s2;
}

// Edge pass 1: e = lrelu(el[src]+er[dst]); segment max into emax[dst]
__global__ void edge_att1_kernel(const float* __restrict__ el, const float* __restrict__ er,
                                 const int* __restrict__ src, const int* __restrict__ dst,
                                 float* __restrict__ ebuf, unsigned int* __restrict__ emax) {
    int e = blockIdx.x * blockDim.x + threadIdx.x;
    if (e >= N_EDGES) return;
    int s = src[e], d = dst[e];
    #pragma unroll
    for (int hh = 0; hh < HEADS; ++hh) {
        float v = lrelu(el[s * HEADS + hh] + er[d * HEADS + hh]);
        ebuf[(size_t)e * HEADS + hh] = v;
        atomicMax(&emax[d * HEADS + hh], fenc(v));
    }
}

// Edge pass 2: ez = exp(e - emax[dst]); segment sum into esum[dst]
__global__ void edge_att2_kernel(const int* __restrict__ dst, const unsigned int* __restrict__ emax,
                                 float* __restrict__ ebuf, float* __restrict__ esum) {
    int e = blockIdx.x * blockDim.x + threadIdx.x;
    if (e >= N_EDGES) return;
    int d = dst[e];
    #pragma unroll
    for (int hh = 0; hh < HEADS; ++hh) {
        float mx = fdec(emax[d * HEADS + hh]);
        float z = __expf(ebuf[(size_t)e * HEADS + hh] - mx);
        ebuf[(size_t)e * HEADS + hh] = z;
        atomicAdd(&esum[d * HEADS + hh], z);
    }
}

// Edge pass 3: h[dst] += sum_h fh[src,h,:]*alpha[h]; wave per edge, lane = out chan
__global__ __launch_bounds__(256) void edge_att3_kernel(const float* __restrict__ FH,
                                                        const float* __restrict__ ebuf,
                                                        const float* __restrict__ esum,
                                                        const int* __restrict__ src,
                                                        const int* __restrict__ dst,
                                                        float* __restrict__ hbuf) {
    int e = blockIdx.x * 8 + (threadIdx.x >> 5);
    int o = threadIdx.x & 31;
    if (e >= N_EDGES) return;
    int s = src[e], d = dst[e];
    float acc = 0.f;
    #pragma unroll
    for (int hh = 0; hh < HEADS; ++hh) {
        float alpha = ebuf[(size_t)e * HEADS + hh] / esum[d * HEADS + hh];
        acc += FH[(size_t)s * 128 + hh * N_HID + o] * alpha;
    }
    atomicAdd(&hbuf[(size_t)d * N_HID + o], acc);
}

// ---------------------------------------------------------------------------
// BML edge message. Weights staged in LDS (~41KB). Thread per edge.
//   coef[m] = lrelu(hd.Wc[r,m,:] + bc[r,m]);  w2[i] = hs.Wc2[r,i,:]
//   res[o]  = sum_{i<8,m<8} Ww[r,o,i,m] * w2[i] * coef[m]   (label-consistent i)
// ---------------------------------------------------------------------------
__global__ __launch_bounds__(256) void bml_kernel(const float* __restrict__ H,
                                                  const float* __restrict__ Wc,
                                                  const float* __restrict__ bc,
                                                  const float* __restrict__ Wc2,
                                                  const float* __restrict__ Ww,
                                                  const int* __restrict__ src,
                                                  const int* __restrict__ dst,
                                                  const int* __restrict__ etype,
                                                  float* __restrict__ agg) {
    __shared__ float sWw[N_RELS * 32 * 8 * 8];   // [r][o][i<8][m] : 32KB
    __shared__ float sWc[N_RELS * MEMD * N_HID]; // 4KB
    __shared__ float sWc2[N_RELS * MEMD * N_HID];
    __shared__ float sbc[N_RELS * MEMD];
    for (int i = threadIdx.x; i < N_RELS * 32 * 64; i += blockDim.x) {
        int r = i >> 11;            // / 2048
        int rem = i & 2047;
        int o = rem >> 6;           // / 64
        int im = rem & 63;
        int ii = im >> 3, mm = im & 7;
        sWw[i] = Ww[(size_t)(((r * 32 + o) * 32 + ii) * 8 + mm)];
    }
    for (int i = threadIdx.x; i < N_RELS * MEMD * N_HID; i += blockDim.x) {
        sWc[i] = Wc[i]; sWc2[i] = Wc2[i];
    }
    if (threadIdx.x < N_RELS * MEMD) sbc[threadIdx.x] = bc[threadIdx.x];
    __syncthreads();

    int e = blockIdx.x * blockDim.x + threadIdx.x;
    if (e >= N_EDGES) return;
    int r = etype[e], s = src[e], d = dst[e];
    float hdv[N_HID], hsv[N_HID];
    const float* hd = H + (size_t)d * N_HID;
    const float* hs = H + (size_t)s * N_HID;
    #pragma unroll
    for (int k = 0; k < N_HID; ++k) { hdv[k] = hd[k]; hsv[k] = hs[k]; }

    float coef[MEMD], w2[MEMD];
    #pragma unroll
    for (int m = 0; m < MEMD; ++m) {
        float a = sbc[r * MEMD + m], b2 = 0.f;
        const float* wc  = &sWc [(r * MEMD + m) * N_HID];
        const float* wc2 = &sWc2[(r * MEMD + m) * N_HID];
        #pragma unroll
        for (int k = 0; k < N_HID; ++k) { a += hdv[k] * wc[k]; b2 += hsv[k] * wc2[k]; }
        coef[m] = lrelu(a);
        w2[m] = b2;
    }
    float* ag = agg + (size_t)d * N_HID;
    const float* wwr = &sWw[r * 2048];
    for (int o = 0; o < N_HID; ++o) {
        const float* wo = wwr + o * 64;
        float acc = 0.f;
        #pragma unroll
        for (int i2 = 0; i2 < MEMD; ++i2) {
            float t2 = 0.f;
            #pragma unroll
            for (int mm = 0; mm < MEMD; ++mm) t2 += wo[i2 * 8 + mm] * coef[mm];
            acc += t2 * w2[i2];
        }
        atomicAdd(&ag[o], acc);
    }
}

// x_new = lrelu(agg/max(deg,1) + h_bias); write into x and xcat slot
__global__ void node_update_kernel(const float* __restrict__ agg, const float* __restrict__ deg,
                                   const float* __restrict__ hb, float* __restrict__ x,
                                   float* __restrict__ xcat, int layer) {
    int t = blockIdx.x * blockDim.x + threadIdx.x;
    if (t >= N_NODES * N_HID) return;
    int n = t >> 5, o = t & 31;
    float v = agg[t] / fmaxf(deg[n], 1.f) + hb[o];
    v = lrelu(v);
    x[t] = v;
    xcat[(size_t)n * 96 + (layer + 1) * N_HID + o] = v;
}

// LayerNorm over last dim (96) with gain/bias
__global__ void layernorm_kernel(const float* __restrict__ xcat, const float* __restrict__ g,
                                 const float* __restrict__ b, float* __restrict__ out) {
    int n = blockIdx.x * blockDim.x + threadIdx.x;
    if (n >= N_NODES) return;
    const float* row = xcat + (size_t)n * 96;
    float s = 0.f, ss = 0.f;
    for (int j = 0; j < 96; ++j) { float v = row[j]; s += v; ss += v * v; }
    float mu = s * (1.f / 96.f);
    float var = ss * (1.f / 96.f) - mu * mu;
    float inv = rsqrtf(var + 1e-5f);
    float* orow = out + (size_t)n * 96;
    for (int j = 0; j < 96; ++j) orow[j] = (row[j] - mu) * inv * g[j] + b[j];
}

// ---------------------------------------------------------------------------
extern "C" void kernel_launch(void* const* d_in, const int* in_sizes, int n_in,
                              void* d_out, int out_size, void* d_ws, size_t ws_size,
                              hipStream_t stream) {
    const float* emb    = (const float*)d_in[0];
    const float* gat_W  = (const float*)d_in[1];   // [2,2,32,128]
    const float* gat_al = (const float*)d_in[2];   // [2,2,4,32]
    const float* gat_ar = (const float*)d_in[3];
    const float* gat_b  = (const float*)d_in[4];   // [2,2,128]
    const float* bml_Wc = (const float*)d_in[5];   // [2,4,8,32]
    const float* bml_bc = (const float*)d_in[6];   // [2,4,8]
    const float* bml_Wc2= (const float*)d_in[7];   // [2,4,8,32]
    const float* bml_Ww = (const float*)d_in[8];   // [2,4,32,32,8]
    const float* h_bias = (const float*)d_in[9];   // [2,32]
    const float* ln_g   = (const float*)d_in[10];  // [96]
    const float* ln_b   = (const float*)d_in[11];  // [96]
    const int*   src    = (const int*)d_in[13];
    const int*   dst    = (const int*)d_in[14];
    const int*   etype  = (const int*)d_in[15];
    float* out = (float*)d_out;

    // workspace carve-up (bytes)
    char* ws = (char*)d_ws;
    size_t off = 0;
    auto carve = [&](size_t nbytes) { char* p = ws + off; off += (nbytes + 255) & ~(size_t)255; return (void*)p; };
    float*        xcat = (float*)carve((size_t)N_NODES * 96 * 4);
    float*        x    = (float*)carve((size_t)N_NODES * 32 * 4);
    float*        fh   = (float*)carve((size_t)N_NODES * 128 * 4);
    float*        el   = (float*)carve((size_t)N_NODES * 4 * 4);
    float*        er   = (float*)carve((size_t)N_NODES * 4 * 4);
    unsigned int* emax = (unsigned int*)carve((size_t)N_NODES * 4 * 4);
    float*        esum = (float*)carve((size_t)N_NODES * 4 * 4);
    float*        hbuf = (float*)carve((size_t)N_NODES * 32 * 4);
    float*        agg  = (float*)carve((size_t)N_NODES * 32 * 4);
    float*        deg  = (float*)carve((size_t)N_NODES * 4);
    float*        ebuf = (float*)carve((size_t)N_EDGES * 4 * 4);
    (void)ws_size; (void)in_sizes; (void)n_in; (void)out_size;

    const int NTHREADS = 256;
    const int g_nw   = (N_NODES * 32) / NTHREADS;        // 6250
    const int g_edge = N_EDGES / NTHREADS;               // 3125
    const int g_n4   = (N_NODES * 4 + NTHREADS - 1) / NTHREADS;
    const int g_node = (N_NODES + NTHREADS - 1) / NTHREADS;

    global_init_kernel<<<g_nw, NTHREADS, 0, stream>>>(emb, xcat, x, deg);
    deg_kernel<<<g_edge, NTHREADS, 0, stream>>>(dst, deg);

    for (int l = 0; l < 2; ++l) {
        // Only the last GAT sub-layer (g=1) survives in the reference loop.
        const float* W  = gat_W  + (size_t)(l * 2 + 1) * 32 * 128;
        const float* al = gat_al + (size_t)(l * 2 + 1) * 4 * 32;
        const float* ar = gat_ar + (size_t)(l * 2 + 1) * 4 * 32;
        const float* gb = gat_b  + (size_t)(l * 2 + 1) * 128;
        const float* Wc  = bml_Wc  + (size_t)l * 4 * 8 * 32;
        const float* bc  = bml_bc  + (size_t)l * 4 * 8;
        const float* Wc2 = bml_Wc2 + (size_t)l * 4 * 8 * 32;
        const float* Ww  = bml_Ww  + (size_t)l * 4 * 32 * 32 * 8;
        const float* hb  = h_bias  + (size_t)l * 32;

        layer_init_kernel<<<g_nw, NTHREADS, 0, stream>>>(gb, hbuf, agg, emax, esum);
        gemm_fh_kernel<<<N_NODES / 16, NTHREADS, 0, stream>>>(x, W, fh);
        eler_kernel<<<g_n4, NTHREADS, 0, stream>>>(fh, al, ar, el, er);
        edge_att1_kernel<<<g_edge, NTHREADS, 0, stream>>>(el, er, src, dst, ebuf, emax);
        edge_att2_kernel<<<g_edge, NTHREADS, 0, stream>>>(dst, emax, ebuf, esum);
        edge_att3_kernel<<<N_EDGES / 8, NTHREADS, 0, stream>>>(fh, ebuf, esum, src, dst, hbuf);
        bml_kernel<<<g_edge, NTHREADS, 0, stream>>>(hbuf, Wc, bc, Wc2, Ww, src, dst, etype, agg);
        node_update_kernel<<<g_nw, NTHREADS, 0, stream>>>(agg, deg, hb, x, xcat, l);
    }
    layernorm_kernel<<<g_node, NTHREADS, 0, stream>>>(xcat, ln_g, ln_b, out);
}